// DepthEstimationNet_27109833572601
// MI455X (gfx1250) — compile-verified
//
#include <hip/hip_runtime.h>

// CDNA5 / gfx1250 kernel for:
//   sparse_m[b,i,j] = d[b,i] * (1/d[b,j]);  idx = searchsorted(q, sparse_m, right);  out = inv[idx]
// B=8, S=48*48=2304, 40 thresholds, 41 codebook entries.
//
// Design (store-bandwidth bound: 170MB out / 23.3 TB/s ~ 7.3us floor):
//  - V_WMMA_F32_16X16X4_F32 computes each 16x16 ratio tile as a rank-1 GEMM
//    (A: 16x4 with d_i in K=0; B: 4x16 with 1/d_j in K=0).
//  - Branchless upper-bound binary search (6 steps) in a 64-entry LDS table
//    (q padded with +INF). 64 dwords => every index hits a distinct LDS bank,
//    so the data-dependent ds_load_b32 per step is conflict-free. Same for the
//    41-entry inv codebook gather.
//  - WMMA C/D layout (VGPR r -> rows r, r+8; lanes -> 16 consecutive cols)
//    maps to coalesced row-major stores (two 64B segments per store).

typedef float v2f __attribute__((ext_vector_type(2)));
typedef float v8f __attribute__((ext_vector_type(8)));

#define NB 40
#define S 2304
#define TILE_COLS 256

__global__ __launch_bounds__(256)
void lloyd_relmap_kernel(const float* __restrict__ x,
                         const float* __restrict__ q,
                         const float* __restrict__ invc,
                         float* __restrict__ out) {
  __shared__ float qpad[64];     // thresholds padded to 64 with +INF (bank-conflict-free)
  __shared__ float invt[64];     // 41 codebook entries, padded (bank-conflict-free)
  __shared__ float drow[16];     // d values for this row tile
  __shared__ float rcol[TILE_COLS]; // 1/d for this column tile

  const int tid   = threadIdx.x;
  const int ctile = blockIdx.x;  // 0..8   (cols of 256)
  const int rtile = blockIdx.y;  // 0..143 (rows of 16)
  const int b     = blockIdx.z;  // 0..7

  const int r0 = rtile * 16;
  const int c0 = ctile * TILE_COLS;
  const float* db = x + (size_t)b * S;

  // Stage tables + tile operands into LDS.
  if (tid < 64) {
    qpad[tid] = (tid < NB) ? q[tid] : __builtin_inff();
  } else if (tid < 128) {
    const int k = tid - 64;
    invt[k] = (k <= NB) ? invc[k] : 0.0f;
  } else if (tid < 144) {
    drow[tid - 128] = db[r0 + (tid - 128)];
  }
  rcol[tid] = 1.0f / db[c0 + tid];   // matches reference: d * (1.0/d)
  __syncthreads();

  const int wave     = tid >> 5;     // 0..7: each wave owns a 16x32 strip
  const int lane     = tid & 31;
  const int laneHalf = lane >> 4;    // 0 -> rows 0..7, 1 -> rows 8..15 (C/D layout)
  const int lane16   = lane & 15;

  // A matrix (16x4 f32): VGPR0 = {lanes0-15: K=0 col = d_i, lanes16-31: K=2 = 0}
  //                      VGPR1 = {K=1, K=3} = 0
  v2f a;
  a.x = (laneHalf == 0) ? drow[lane16] : 0.0f;
  a.y = 0.0f;

  float* orow = out + ((size_t)b * S + (size_t)(r0 + laneHalf * 8)) * S + c0;

#pragma unroll
  for (int t = 0; t < 2; ++t) {
    const int ncol = wave * 32 + t * 16;  // local column of this 16x16 subtile

    // B matrix (4x16 f32): VGPR0 = {lanes0-15: K=0 row = 1/d_j, lanes16-31: K=2 = 0}
    v2f bb;
    bb.x = (laneHalf == 0) ? rcol[ncol + lane16] : 0.0f;
    bb.y = 0.0f;

    v8f c = {};
    // D[m][n] = d_m * (1/d_n)   (rank-1 product through the matrix core)
    v8f d = __builtin_amdgcn_wmma_f32_16x16x4_f32(
        /*neg_a=*/false, a, /*neg_b=*/false, bb,
        /*c_mod=*/(short)0, c, /*reuse_a=*/false, /*reuse_b=*/false);

#pragma unroll
    for (int r = 0; r < 8; ++r) {
      const float val = d[r];
      // Branchless upper_bound over 64 sorted entries -> count of q_k <= val.
      // Max reachable count with finite val is 40 (pads are +INF).
      unsigned idx = 0;
      idx += (qpad[idx + 31] <= val) ? 32u : 0u;
      idx += (qpad[idx + 15] <= val) ? 16u : 0u;
      idx += (qpad[idx + 7]  <= val) ?  8u : 0u;
      idx += (qpad[idx + 3]  <= val) ?  4u : 0u;
      idx += (qpad[idx + 1]  <= val) ?  2u : 0u;
      idx += (qpad[idx]      <= val) ?  1u : 0u;
      // Codebook gather + coalesced store (two 64B segments per instruction).
      orow[(size_t)r * S + ncol + lane16] = invt[idx];
    }
  }
}

extern "C" void kernel_launch(void* const* d_in, const int* in_sizes, int n_in,
                              void* d_out, int out_size, void* d_ws, size_t ws_size,
                              hipStream_t stream) {
  const float* x    = (const float*)d_in[0]; // [8*2304] depths
  const float* q    = (const float*)d_in[1]; // [40] sorted thresholds
  const float* invc = (const float*)d_in[2]; // [41] codebook
  float* out = (float*)d_out;                // [8*2304*2304]

  dim3 grid(S / TILE_COLS, S / 16, 8);       // 9 x 144 x 8 blocks
  lloyd_relmap_kernel<<<grid, 256, 0, stream>>>(x, q, invc, out);
}